// LSTM_2044404433067
// MI455X (gfx1250) — compile-verified
//
#include <hip/hip_runtime.h>
#include <hip/hip_bf16.h>
#include <cmath>
#include <cstdint>

#define SEQ   512
#define BATCH 64
#define INDIM 1024
#define HDIM  1024
#define GDIM  (4 * HDIM)

typedef __attribute__((ext_vector_type(2))) float v2f;
typedef __attribute__((ext_vector_type(8))) float v8f;

#define WMMA_F32(Afrag, Bfrag, Cacc) \
  __builtin_amdgcn_wmma_f32_16x16x4_f32(false, (Afrag), false, (Bfrag), \
                                        (short)0, (Cacc), false, false)

// ---------------------------------------------------------------------------
// Async copy one 16B chunk per lane into LDS (ASYNCcnt-tracked, CDNA5).
// VDST VGPR holds the LDS byte offset (low 32 bits of a generic LDS pointer).
// ---------------------------------------------------------------------------
__device__ __forceinline__ void async_b128(uint32_t lds_off, const float* gptr) {
  asm volatile("global_load_async_to_lds_b128 %0, %1, off"
               :: "v"(lds_off), "v"((uint64_t)(uintptr_t)gptr)
               : "memory");
}
__device__ __forceinline__ void wait_async0() {
  asm volatile("s_wait_asynccnt 0" ::: "memory");
}

// ---------------------------------------------------------------------------
// C[M,N] = A[M,K] @ W[N,K]^T + bias0[n] + bias1[n]
// One wave per 32x32 C block = 2x2 WMMA tiles: each A/B fragment feeds two
// v_wmma_f32_16x16x4_f32 -> load:WMMA ratio 1:1 instead of 2:1.
// ---------------------------------------------------------------------------
__global__ __launch_bounds__(256) void wmma_gemm_bias(
    const float* __restrict__ A, const float* __restrict__ W,
    const float* __restrict__ bias0, const float* __restrict__ bias1,
    float* __restrict__ C, int M, int N, int K)
{
  const int lane = threadIdx.x & 31;
  const int wid  = blockIdx.x * (blockDim.x >> 5) + (threadIdx.x >> 5);
  const int ng   = N >> 5;                 // 32-wide groups along N
  const int gm   = wid / ng;
  const int gn   = wid - gm * ng;
  if (gm * 32 >= M) return;

  const int l15 = lane & 15;
  const int hi  = lane >> 4;               // 0 or 1
  const int kk  = hi * 2;                  // K sub-offset 0 or 2
  const float* Ap0 = A + (size_t)(gm * 32 +      l15) * K + kk;
  const float* Ap1 = A + (size_t)(gm * 32 + 16 + l15) * K + kk;
  const float* Wp0 = W + (size_t)(gn * 32 +      l15) * K + kk;
  const float* Wp1 = W + (size_t)(gn * 32 + 16 + l15) * K + kk;

  v8f c00 = {}, c01 = {}, c10 = {}, c11 = {};
  #pragma unroll 4
  for (int k = 0; k < K; k += 4) {
    v2f a0 = *(const v2f*)(Ap0 + k);
    v2f a1 = *(const v2f*)(Ap1 + k);
    v2f b0 = *(const v2f*)(Wp0 + k);
    v2f b1 = *(const v2f*)(Wp1 + k);
    c00 = WMMA_F32(a0, b0, c00);
    c01 = WMMA_F32(a0, b1, c01);
    c10 = WMMA_F32(a1, b0, c10);
    c11 = WMMA_F32(a1, b1, c11);
  }

  const int col0 = gn * 32 + l15;
  const int col1 = col0 + 16;
  const float bs0 = bias0[col0] + bias1[col0];
  const float bs1 = bias0[col1] + bias1[col1];
  const int rm0 = gm * 32 + hi * 8;        // C layout: row = reg + 8*(lane>=16)
  const int rm1 = rm0 + 16;
  #pragma unroll
  for (int r = 0; r < 8; ++r) {
    C[(size_t)(rm0 + r) * N + col0] = c00[r] + bs0;
    C[(size_t)(rm0 + r) * N + col1] = c01[r] + bs1;
    C[(size_t)(rm1 + r) * N + col0] = c10[r] + bs0;
    C[(size_t)(rm1 + r) * N + col1] = c11[r] + bs1;
  }
}

// ---------------------------------------------------------------------------
// One recurrence timestep: z = Zx[t] + h_{t-1} @ Wh^T ; gates ; c,h update.
// grid = (HDIM/16, 2); block = 128 = 4 waves (one per gate i,f,g,o).
// Each wave computes 2 batch tiles (32 rows) for its 16 h-columns, sharing
// one Wh fragment across both WMMAs. h_{t-1} K-panels are double-buffered in
// LDS via GLOBAL_LOAD_ASYNC_TO_LDS_B128: panel p+1 streams while panel p
// computes; s_wait_asynccnt 0 + one barrier per panel publishes the data.
// ---------------------------------------------------------------------------
__global__ __launch_bounds__(128) void lstm_step(
    const float* __restrict__ Zx,    // [BATCH, 4H] for this timestep
    const float* __restrict__ Wh,    // [4H, H]
    const float* __restrict__ h_in,  // [BATCH, H]  (h_{t-1})
    float* __restrict__ c_state,     // [BATCH, H]  (block-exclusive region)
    float* __restrict__ h_out)       // [BATCH, H]  (h_t)
{
  __shared__ float Alds[2][32][132];  // 2 x (32 rows x 128 K-panel), +4 pad
  __shared__ float zt[4][32][16];     // i,f,g,o tiles for 32 batch rows

  const int tn   = blockIdx.x;        // h-column tile 0..63
  const int mh   = blockIdx.y;        // batch half   0..1
  const int tid  = threadIdx.x;
  const int wave = tid >> 5;          // gate index 0..3
  const int lane = tid & 31;
  const int l15  = lane & 15;
  const int hi   = lane >> 4;
  const int kk   = hi * 2;
  const int rowbase = mh * 32;

  const float* hsrc = h_in + (size_t)rowbase * HDIM;
  const float* Wp   = Wh + (size_t)(wave * HDIM + tn * 16 + l15) * HDIM + kk;
  const uint32_t lds0 = (uint32_t)(uintptr_t)&Alds[0][0][0];
  const uint32_t lds1 = (uint32_t)(uintptr_t)&Alds[1][0][0];

  // issue panel 0 (32 rows x 128 floats = 1024 x b128, 8 per thread)
  for (int e = tid; e < 32 * 32; e += 128) {
    const int r = e >> 5, c4 = e & 31;
    async_b128(lds0 + (uint32_t)((r * 132 + c4 * 4) * 4),
               hsrc + (size_t)r * HDIM + c4 * 4);
  }

  v8f acc0 = {}, acc1 = {};
  const int NP = HDIM / 128;          // 8 panels
  for (int p = 0; p < NP; ++p) {
    wait_async0();                    // my panel-p copies have landed
    __syncthreads();                  // everyone's have landed
    if (p + 1 < NP) {                 // prefetch next panel into other buffer
      const uint32_t ldsn = ((p + 1) & 1) ? lds1 : lds0;
      const int kp = (p + 1) * 128;
      for (int e = tid; e < 32 * 32; e += 128) {
        const int r = e >> 5, c4 = e & 31;
        async_b128(ldsn + (uint32_t)((r * 132 + c4 * 4) * 4),
                   hsrc + (size_t)r * HDIM + kp + c4 * 4);
      }
    }
    const int buf = p & 1;
    const float* Ap0 = &Alds[buf][l15][kk];
    const float* Ap1 = &Alds[buf][16 + l15][kk];
    const float* Bp  = Wp + p * 128;
    #pragma unroll 8
    for (int k = 0; k < 128; k += 4) {
      v2f b  = *(const v2f*)(Bp + k);      // one Wh fragment ...
      v2f a0 = *(const v2f*)(Ap0 + k);
      v2f a1 = *(const v2f*)(Ap1 + k);
      acc0 = WMMA_F32(a0, b, acc0);        // ... feeds two WMMAs
      acc1 = WMMA_F32(a1, b, acc1);
    }
  }

  // add precomputed input projection, park gate tiles in LDS
  const int zcol = wave * HDIM + tn * 16 + l15;
  #pragma unroll
  for (int r = 0; r < 8; ++r) {
    const int rb = hi * 8 + r;
    zt[wave][rb][l15] =
        acc0[r] + Zx[(size_t)(rowbase + rb) * GDIM + zcol];
    zt[wave][16 + rb][l15] =
        acc1[r] + Zx[(size_t)(rowbase + 16 + rb) * GDIM + zcol];
  }
  __syncthreads();

  // fused gate nonlinearity + state update (512 elems over 128 threads)
  for (int e = tid; e < 512; e += 128) {
    const int r = e >> 4, c = e & 15;
    const float zi = zt[0][r][c], zf = zt[1][r][c];
    const float zg = zt[2][r][c], zo = zt[3][r][c];
    const size_t idx = (size_t)(rowbase + r) * HDIM + tn * 16 + c;
    const float cp = c_state[idx];
    const float si = 1.0f / (1.0f + __expf(-zi));
    const float sf = 1.0f / (1.0f + __expf(-zf));
    const float so = 1.0f / (1.0f + __expf(-zo));
    const float cn = sf * cp + si * tanhf(zg);
    c_state[idx] = cn;
    h_out[idx]   = so * tanhf(cn);
  }
}

// ---------------------------------------------------------------------------
extern "C" void kernel_launch(void* const* d_in, const int* in_sizes, int n_in,
                              void* d_out, int out_size, void* d_ws, size_t ws_size,
                              hipStream_t stream)
{
  (void)in_sizes; (void)n_in; (void)out_size;
  const float* x   = (const float*)d_in[0];
  const float* Wx0 = (const float*)d_in[1];
  const float* bx0 = (const float*)d_in[2];
  const float* Wh0 = (const float*)d_in[3];
  const float* bh0 = (const float*)d_in[4];
  const float* Wx1 = (const float*)d_in[5];
  const float* bx1 = (const float*)d_in[6];
  const float* Wh1 = (const float*)d_in[7];
  const float* bh1 = (const float*)d_in[8];
  float* out = (float*)d_out;                    // [SEQ, BATCH, HDIM]

  // chunk size: largest divisor of SEQ whose scratch fits in ws
  int CH = 1;
  static const int cands[10] = {512, 256, 128, 64, 32, 16, 8, 4, 2, 1};
  for (int i = 0; i < 10; ++i) {
    const size_t need =
        ((size_t)cands[i] * (2ull * BATCH * GDIM + (size_t)BATCH * HDIM) +
         4ull * BATCH * HDIM) * sizeof(float);
    if (need <= ws_size) { CH = cands[i]; break; }
  }

  float* zx0   = (float*)d_ws;                               // [CH, BATCH, 4H]
  float* zx1   = zx0 + (size_t)CH * BATCH * GDIM;            // [CH, BATCH, 4H]
  float* hseq0 = zx1 + (size_t)CH * BATCH * GDIM;            // [CH+1, BATCH, H], slot0 = carry
  float* c0    = hseq0 + (size_t)(CH + 1) * BATCH * HDIM;    // [BATCH, H]
  float* c1    = c0 + (size_t)BATCH * HDIM;                  // [BATCH, H]
  float* hzero = c1 + (size_t)BATCH * HDIM;                  // [BATCH, H]

  const size_t stateB = (size_t)BATCH * HDIM * sizeof(float);
  hipMemsetAsync(c0,    0, stateB, stream);
  hipMemsetAsync(c1,    0, stateB, stream);
  hipMemsetAsync(hzero, 0, stateB, stream);
  hipMemsetAsync(hseq0, 0, stateB, stream);    // layer-0 h carry-in

  const dim3 stepGrid(HDIM / 16, 2);
  const int nchunks = SEQ / CH;
  for (int ch = 0; ch < nchunks; ++ch) {
    const int t0 = ch * CH;
    const int M  = CH * BATCH;
    const int groups = (M / 32) * (GDIM / 32);
    const int gemmBlocks = (groups + 7) / 8;

    // layer 0 input projection for this chunk (bias folded: bx0+bh0)
    wmma_gemm_bias<<<gemmBlocks, 256, 0, stream>>>(
        x + (size_t)t0 * BATCH * INDIM, Wx0, bx0, bh0, zx0, M, GDIM, INDIM);

    // layer 0 recurrence
    for (int t = 0; t < CH; ++t)
      lstm_step<<<stepGrid, 128, 0, stream>>>(
          zx0 + (size_t)t * BATCH * GDIM, Wh0,
          hseq0 + (size_t)t * BATCH * HDIM, c0,
          hseq0 + (size_t)(t + 1) * BATCH * HDIM);

    // carry h0 for next chunk
    hipMemcpyAsync(hseq0, hseq0 + (size_t)CH * BATCH * HDIM, stateB,
                   hipMemcpyDeviceToDevice, stream);

    // layer 1 input projection from this chunk's layer-0 outputs
    wmma_gemm_bias<<<gemmBlocks, 256, 0, stream>>>(
        hseq0 + (size_t)BATCH * HDIM, Wx1, bx1, bh1, zx1, M, GDIM, HDIM);

    // layer 1 recurrence, writing straight into d_out (doubles as h-history)
    for (int t = 0; t < CH; ++t) {
      const int tg = t0 + t;
      const float* hin = (tg == 0) ? hzero : out + (size_t)(tg - 1) * BATCH * HDIM;
      lstm_step<<<stepGrid, 128, 0, stream>>>(
          zx1 + (size_t)t * BATCH * GDIM, Wh1, hin, c1,
          out + (size_t)tg * BATCH * HDIM);
    }
  }
}